// Detector_29686813950335
// MI455X (gfx1250) — compile-verified
//
#include <hip/hip_runtime.h>
#include <hip/hip_bf16.h>
#include <stdint.h>

#define BB    4
#define NN    6300
#define NC    80
#define MM    8192          // N padded to power of two for bitonic sort
#define CONF  0.5f
#define NMST  0.4f

// ---------------- gfx1250 async global->LDS path (probe via __has_builtin) ---
#if defined(__has_builtin)
#  if __has_builtin(__builtin_amdgcn_global_load_async_to_lds_b128)
#    define HAVE_ASYNC_LD 1
#  endif
#  if __has_builtin(__builtin_amdgcn_global_store_async_from_lds_b128)
#    define HAVE_ASYNC_ST 1
#  endif
#  if __has_builtin(__builtin_amdgcn_s_wait_asynccnt)
#    define HAVE_WAIT_ASYNC 1
#  endif
#endif

typedef int v4i __attribute__((vector_size(16)));   // matches builtin param pointee

__device__ __forceinline__ void async_copy16_g2l(const void* g, void* l) {
#if defined(HAVE_ASYNC_LD)
  __builtin_amdgcn_global_load_async_to_lds_b128(
      (__attribute__((address_space(1))) v4i*)(uintptr_t)g,
      (__attribute__((address_space(3))) v4i*)l, 0, 0);
#else
  *(float4*)l = *(const float4*)g;
#endif
}

__device__ __forceinline__ void async_copy16_l2g(void* g, const void* l) {
#if defined(HAVE_ASYNC_ST)
  __builtin_amdgcn_global_store_async_from_lds_b128(
      (__attribute__((address_space(1))) v4i*)g,
      (__attribute__((address_space(3))) v4i*)(uintptr_t)l, 0, 0);
#else
  *(float4*)g = *(const float4*)l;
#endif
}

__device__ __forceinline__ void async_wait0() {
#if defined(HAVE_WAIT_ASYNC)
  __builtin_amdgcn_s_wait_asynccnt(0);
#elif defined(HAVE_ASYNC_LD) || defined(HAVE_ASYNC_ST)
  asm volatile("s_wait_asynccnt 0" ::: "memory");
#endif
}

// ---------------- Kernel A: decode, bbox_fit, det output, sort keys ----------
__global__ void __launch_bounds__(256)
prep_kernel(const float* __restrict__ pred,
            const int* __restrict__ fhp, const int* __restrict__ fwp,
            float* __restrict__ det_out,
            unsigned long long* __restrict__ keys,
            float4* __restrict__ oboxes)
{
  int t = blockIdx.x * blockDim.x + threadIdx.x;
  int b = t >> 13, n = t & (MM - 1);              // MM = 8192
  if (b >= BB) return;
  unsigned long long* kb = keys + (size_t)b * MM;
  if (n >= NN) { kb[n] = ~0ull; return; }         // padding sorts last

  const float* p = pred + ((size_t)b * NN + n) * (5 + NC);
  float cx = p[0], cy = p[1], w = p[2], h = p[3], obj = p[4];

  // argmax/max over classes (first occurrence wins, matches jnp.argmax)
  float best = p[5]; int bi = 0;
  #pragma unroll 4
  for (int c = 1; c < NC; ++c) {
    float v = p[5 + c];
    if (v > best) { best = v; bi = c; }
  }

  float x1 = cx - w * 0.5f, y1 = cy - h * 0.5f;
  float x2 = cx + w * 0.5f, y2 = cy + h * 0.5f;

  // bbox_fit (fh/fw are runtime scalars)
  float fh = (float)fhp[0], fw = (float)fwp[0];
  const float ih = 320.f, iw = 320.f;
  float fdim  = fmaxf(fh, fw);
  float pad_x = fmaxf((fh * iw / ih - fw) * 0.5f, 0.f) * iw / fdim;
  float pad_y = fmaxf((fw * ih / iw - fh) * 0.5f, 0.f) * ih / fdim;
  float sx = fw / (iw - 2.f * pad_x);
  float sy = fh / (ih - 2.f * pad_y);

  float* d = det_out + ((size_t)b * NN + n) * 7;
  d[0] = fminf(fmaxf((x1 - pad_x) * sx, 0.f), fw);
  d[1] = fminf(fmaxf((y1 - pad_y) * sy, 0.f), fh);
  d[2] = fminf(fmaxf((x2 - pad_x) * sx, 0.f), fw);
  d[3] = fminf(fmaxf((y2 - pad_y) * sy, 0.f), fh);
  d[4] = obj; d[5] = best; d[6] = (float)bi;

  float off = (float)bi * 10000.f;                 // class-offset boxes for IoU
  oboxes[(size_t)b * NN + n] = make_float4(x1 + off, y1 + off, x2 + off, y2 + off);

  // sort key: descending score (valid ? obj : -inf), stable tie-break by index
  bool  valid = (obj >= CONF);
  float score = valid ? obj : -__builtin_inff();
  unsigned u = __float_as_uint(score);
  u ^= (u & 0x80000000u) ? 0xFFFFFFFFu : 0x80000000u;  // float -> ascending uint
  unsigned d32 = ~u;                                    // invert -> descending
  kb[n] = ((unsigned long long)d32 << 32) | (unsigned)n;
}

// ---------------- Kernel B: per-image bitonic sort of 8192 u64 keys in LDS ---
__global__ void __launch_bounds__(1024)
sort_kernel(unsigned long long* __restrict__ keys)
{
  extern __shared__ unsigned long long sk[];       // MM entries = 64 KB
  unsigned long long* kb = keys + (size_t)blockIdx.x * MM;
  const int tid = threadIdx.x;

  for (int i = tid; i < MM / 2; i += 1024)         // 16B async stages into LDS
    async_copy16_g2l(kb + 2 * i, sk + 2 * i);
  async_wait0();
  __syncthreads();

  for (int k = 2; k <= MM; k <<= 1) {
    for (int j = k >> 1; j > 0; j >>= 1) {
      #pragma unroll
      for (int r = 0; r < MM / 1024; ++r) {        // each thread owns 8 slots
        int i = r * 1024 + tid;
        int l = i ^ j;
        if (l > i) {
          unsigned long long a = sk[i], c = sk[l];
          bool asc = ((i & k) == 0);
          if ((a > c) == asc) { sk[i] = c; sk[l] = a; }
        }
      }
      __syncthreads();
    }
  }

  for (int i = tid; i < MM / 2; i += 1024)
    async_copy16_l2g(kb + 2 * i, sk + 2 * i);
  async_wait0();
}

// ---------------- Kernel C: LDS-resident greedy NMS + keep scatter -----------
__global__ void __launch_bounds__(1024)
nms_kernel(const unsigned long long* __restrict__ keys,
           const float4* __restrict__ oboxes,
           float* __restrict__ keep_out)
{
  extern __shared__ char smem[];
  float4*   box   = (float4*)smem;                       // NN sorted boxes
  unsigned* flags = (unsigned*)(smem + (size_t)NN * 16); // bit0=valid bit1=sup bit2=keep
  __shared__ unsigned nv;

  const int b = blockIdx.x, tid = threadIdx.x;
  const unsigned long long* kb = keys + (size_t)b * MM;
  if (tid == 0) nv = 0;

  // gather boxes in sorted order straight into LDS (per-lane async b128)
  for (int i = tid; i < NN; i += 1024) {
    unsigned long long key = kb[i];
    unsigned orig = (unsigned)key;
    flags[i] = ((unsigned)(key >> 32) < 0xFF800000u) ? 1u : 0u;  // valid prefix
    async_copy16_g2l(&oboxes[(size_t)b * NN + orig], &box[i]);
  }
  async_wait0();
  __syncthreads();

  for (int i = tid; i < NN; i += 1024)
    if (flags[i] & 1u) atomicMax(&nv, (unsigned)(i + 1));
  __syncthreads();
  const int nvalid = (int)nv;   // sorted => valid boxes are exactly [0, nvalid)

  // Greedy NMS. Barrier only after iterations that WRITE flags (kept boxes):
  // between two barriers at most one iteration writes, so flag reads in the
  // non-kept iterations are already coherent. `kept` is workgroup-uniform
  // (derived from LDS state stable since the last barrier), so the
  // conditional __syncthreads is uniform.
  for (int i = 0; i < nvalid; ++i) {
    unsigned fi = flags[i];
    bool kept = (fi & 1u) && !(fi & 2u);
    if (kept) {
      if (tid == 0) flags[i] = fi | 4u;
      float4 bi = box[i];
      float areai = (bi.z - bi.x + 1.f) * (bi.w - bi.y + 1.f);
      for (int j = i + 1 + tid; j < nvalid; j += 1024) {
        float4 bj = box[j];
        float ww = fmaxf(fminf(bi.z, bj.z) - fmaxf(bi.x, bj.x) + 1.f, 0.f);
        float hh = fmaxf(fminf(bi.w, bj.w) - fmaxf(bi.y, bj.y) + 1.f, 0.f);
        float inter = ww * hh;
        float areaj = (bj.z - bj.x + 1.f) * (bj.w - bj.y + 1.f);
        float iou = inter / (areai + areaj - inter + 1e-16f);
        if (iou >= NMST) flags[j] |= 2u;   // unique thread per j this iteration
      }
      __syncthreads();
    }
  }
  __syncthreads();   // last kept iteration's writes -> visible to scatter

  for (int i = tid; i < NN; i += 1024) {
    unsigned orig = (unsigned)kb[i];
    keep_out[(size_t)b * NN + orig] = (flags[i] & 4u) ? 1.0f : 0.0f;
  }
}

// ---------------- launch ------------------------------------------------------
extern "C" void kernel_launch(void* const* d_in, const int* in_sizes, int n_in,
                              void* d_out, int out_size, void* d_ws, size_t ws_size,
                              hipStream_t stream) {
  const float* pred = (const float*)d_in[0];
  const int*   fhp  = (const int*)d_in[1];
  const int*   fwp  = (const int*)d_in[2];

  float* det_out  = (float*)d_out;                       // BB*NN*7 floats
  float* keep_out = (float*)d_out + (size_t)BB * NN * 7; // BB*NN floats

  unsigned long long* keys = (unsigned long long*)d_ws;  // BB*MM*8 = 256 KB
  float4* oboxes = (float4*)((char*)d_ws + (size_t)BB * MM * 8); // BB*NN*16 B

  prep_kernel<<<(BB * MM) / 256, 256, 0, stream>>>(pred, fhp, fwp, det_out, keys, oboxes);
  sort_kernel<<<BB, 1024, (size_t)MM * 8, stream>>>(keys);
  nms_kernel<<<BB, 1024, (size_t)NN * 16 + (size_t)NN * 4, stream>>>(keys, oboxes, keep_out);
}